// AttSampler_86792699118097
// MI455X (gfx1250) — compile-verified
//
#include <hip/hip_runtime.h>
#include <hip/hip_bf16.h>
#include <math.h>

// ---------------------------------------------------------------------------
// AttSampler for MI455X (gfx1250).
//  - data:  (8, 32, 512, 512) f32
//  - attx:  (8, 512, 1) f32     atty: (8, 512, 1) f32
//  - out:   (8, 32, 512, 512) f32   (SCALE = 1.0)
// Memory-bound: ~512 MB traffic -> ~22us floor @ 23.3 TB/s.
// Kernel A builds per-axis (index, weight) tables; Kernel B does the fused
// separable bilinear gather, staging input rows into LDS with the CDNA5
// async global->LDS path (ASYNCcnt).
// ---------------------------------------------------------------------------

#define N_B   8
#define N_C   32
#define SZ    512     // H == W == out_h == out_w == S
#define DENSE 4
#define ITERS 5

#if defined(__AMDGCN__) && __has_builtin(__builtin_amdgcn_global_load_async_to_lds_b128)
#define USE_ASYNC_LDS 1
// Builtin signature (from hipcc diagnostic): (v4i AS1*, v4i AS3*, Ii, Ii)
typedef int v4i __attribute__((vector_size(16)));
typedef __attribute__((address_space(1))) v4i* gptr_v4i;   // global (device) ptr
typedef __attribute__((address_space(3))) v4i* sptr_v4i;   // LDS (shared) ptr
#endif

// ------------------------- Kernel A: coordinate tables ----------------------
// grid = (2 axes, 8 batches), block = 512 threads. Mirrors _axis_coords + the
// coord -> (i0, w) mapping from _lerp_axis (align_corners=True, size=512).
__global__ void att_coords_kernel(const float* __restrict__ attx,
                                  const float* __restrict__ atty,
                                  int*   __restrict__ xi0, float* __restrict__ xw,
                                  int*   __restrict__ yi0, float* __restrict__ yw) {
    __shared__ float a[SZ];     // working / cumsum array
    __shared__ float red[SZ];   // reduction scratch

    const int axis = blockIdx.x;   // 0 = x (width), 1 = y (height)
    const int n    = blockIdx.y;
    const int tid  = threadIdx.x;

    const float* att = (axis == 0 ? attx : atty) + n * SZ;
    float v = att[tid];

    // --- sum(att) ---
    red[tid] = v;
    __syncthreads();
    for (int off = SZ / 2; off > 0; off >>= 1) {
        if (tid < off) red[tid] += red[tid + off];
        __syncthreads();
    }
    float total = red[0];
    __syncthreads();

    v = v / total * (float)SZ;                       // sum == out_size
    const float thr = (float)DENSE * (float)SZ / (float)SZ;  // = 4.0f

    // --- iterative clip + uniform redistribution ---
    for (int it = 0; it < ITERS; ++it) {
        v = fminf(v, thr);
        red[tid] = v;
        __syncthreads();
        for (int off = SZ / 2; off > 0; off >>= 1) {
            if (tid < off) red[tid] += red[tid + off];
            __syncthreads();
        }
        float s = red[0];
        __syncthreads();
        v = v + ((float)SZ - s) / (float)SZ;
    }

    // --- inclusive scan (Hillis-Steele) into a[] ---
    a[tid] = v;
    __syncthreads();
    for (int off = 1; off < SZ; off <<= 1) {
        float add = (tid >= off) ? a[tid - off] : 0.0f;
        __syncthreads();
        a[tid] += add;
        __syncthreads();
    }

    // --- inverse CDF: first j with csum[j] >= tgt ---
    const float step = a[SZ - 1] / (float)SZ;
    const float tgt  = step * (float)(tid + 1);
    int lo = 0, hi = SZ;
    while (lo < hi) {
        int mid = (lo + hi) >> 1;
        if (a[mid] < tgt) lo = mid + 1; else hi = mid;
    }
    int j = lo;
    if (j > SZ - 1) j = SZ - 1;
    const float right = a[j];
    const float left  = (j > 0) ? a[j - 1] : 0.0f;
    float frac = (tgt - left) / fmaxf(right - left, 1e-8f);
    frac = fminf(fmaxf(frac, 0.0f), 1.0f);
    const float coord = ((float)j + frac) / (float)SZ * 2.0f - 1.0f;

    // --- coord -> (i0, w), align_corners=True, size = 512 ---
    const float p   = (coord + 1.0f) * 0.5f * (float)(SZ - 1);
    const float p0f = floorf(p);
    const float w   = p - p0f;
    int i0 = (int)p0f;
    i0 = (i0 < 0) ? 0 : ((i0 > SZ - 1) ? SZ - 1 : i0);

    if (axis == 0) { xi0[n * SZ + tid] = i0; xw[n * SZ + tid] = w; }
    else           { yi0[n * SZ + tid] = i0; yw[n * SZ + tid] = w; }
}

// ------------------------- Kernel B: fused bilinear gather ------------------
// grid = (oy=512, c=32, n=8), block = 512 threads (one output row per block).
// Stage rows y0 and y1 (4 KB) into LDS with async global->LDS b128, then each
// lane gathers its 4 taps from LDS.
__global__ void att_resample_kernel(const float* __restrict__ data,
                                    const int*   __restrict__ xi0,
                                    const float* __restrict__ xw,
                                    const int*   __restrict__ yi0,
                                    const float* __restrict__ yw,
                                    float* __restrict__ out) {
    __shared__ float lds[2 * SZ];   // lds[0..511] = row y0, lds[512..1023] = row y1

    const int ox = threadIdx.x;
    const int oy = blockIdx.x;
    const int c  = blockIdx.y;
    const int n  = blockIdx.z;

    const int   y0 = yi0[n * SZ + oy];
    const float wy = yw [n * SZ + oy];
    const int   y1 = (y0 + 1 < SZ) ? y0 + 1 : SZ - 1;

    const size_t plane = ((size_t)(n * N_C + c)) * (size_t)SZ;
    const float* row0 = data + (plane + (size_t)y0) * (size_t)SZ;
    const float* row1 = data + (plane + (size_t)y1) * (size_t)SZ;

#ifdef USE_ASYNC_LDS
    // 256 lanes x b128 = 4 KB staged asynchronously (ASYNCcnt-tracked).
    if (ox < 256) {
        const int r = ox >> 7;          // 0 -> row0, 1 -> row1
        const int t = ox & 127;         // 16B chunk within the row
        const float* src = (r ? row1 : row0) + (t << 2);
        float* dst = &lds[(r << 9) + (t << 2)];
        __builtin_amdgcn_global_load_async_to_lds_b128(
            (gptr_v4i)src, (sptr_v4i)dst, 0, 0);
    }
#if __has_builtin(__builtin_amdgcn_s_wait_asynccnt)
    __builtin_amdgcn_s_wait_asynccnt(0);
#else
    asm volatile("s_wait_asynccnt 0" ::: "memory");
#endif
    __syncthreads();
#else
    lds[ox]      = row0[ox];
    lds[SZ + ox] = row1[ox];
    __syncthreads();
#endif

    const int   x0 = xi0[n * SZ + ox];
    const float wx = xw [n * SZ + ox];
    const int   x1 = (x0 + 1 < SZ) ? x0 + 1 : SZ - 1;

    // Same FP composition as the reference: x-lerp first, then y-lerp.
    const float top = lds[x0]      * (1.0f - wx) + lds[x1]      * wx;
    const float bot = lds[SZ + x0] * (1.0f - wx) + lds[SZ + x1] * wx;
    out[(plane + (size_t)oy) * (size_t)SZ + (size_t)ox] =
        top * (1.0f - wy) + bot * wy;
}

// ---------------------------------------------------------------------------
extern "C" void kernel_launch(void* const* d_in, const int* in_sizes, int n_in,
                              void* d_out, int out_size, void* d_ws, size_t ws_size,
                              hipStream_t stream) {
    const float* data = (const float*)d_in[0];
    const float* attx = (const float*)d_in[1];
    const float* atty = (const float*)d_in[2];
    float* out = (float*)d_out;

    // Workspace tables: 4 arrays of 8*512 (int or float) = 64 KB total.
    char* ws = (char*)d_ws;
    const size_t tbl = (size_t)N_B * SZ * sizeof(int);   // 16 KB each
    int*   xi0 = (int*)  (ws + 0 * tbl);
    float* xw  = (float*)(ws + 1 * tbl);
    int*   yi0 = (int*)  (ws + 2 * tbl);
    float* yw  = (float*)(ws + 3 * tbl);

    // Kernel A: 2 axes x 8 batches, 512 threads each (tiny).
    att_coords_kernel<<<dim3(2, N_B, 1), dim3(SZ, 1, 1), 0, stream>>>(
        attx, atty, xi0, xw, yi0, yw);

    // Kernel B: one block per output row: (512, 32, 8) x 512 threads.
    att_resample_kernel<<<dim3(SZ, N_C, N_B), dim3(SZ, 1, 1), 0, stream>>>(
        data, xi0, xw, yi0, yw, out);
}